// Model_69020124446838
// MI455X (gfx1250) — compile-verified
//
#include <hip/hip_runtime.h>
#include <math.h>

// Problem constants (from reference)
#define B_   16
#define S_   512
#define V_   8192
#define E_   256
#define H_   512
#define G4H  2048          // 4*H
#define NBLK 16            // persistent blocks in recurrent kernel

typedef __attribute__((ext_vector_type(2))) float v2f;
typedef __attribute__((ext_vector_type(8))) float v8f;

__device__ __forceinline__ float sigm(float x) { return 1.f / (1.f + __expf(-x)); }

#define WMMA_F32(A, Bv, C) \
  __builtin_amdgcn_wmma_f32_16x16x4_f32(false, (A), false, (Bv), (short)0, (C), false, false)

// ---------------------------------------------------------------------------
// Grid barrier (sense via generation counter). bar[0]=arrive cnt, bar[1]=gen.
// ---------------------------------------------------------------------------
__device__ __forceinline__ void grid_barrier(unsigned* cnt, unsigned* gen) {
  __threadfence();
  __syncthreads();
  if (threadIdx.x == 0) {
    unsigned g0 = __hip_atomic_load(gen, __ATOMIC_RELAXED, __HIP_MEMORY_SCOPE_AGENT);
    if (__hip_atomic_fetch_add(cnt, 1u, __ATOMIC_ACQ_REL, __HIP_MEMORY_SCOPE_AGENT) == NBLK - 1u) {
      __hip_atomic_store(cnt, 0u, __ATOMIC_RELAXED, __HIP_MEMORY_SCOPE_AGENT);
      __hip_atomic_fetch_add(gen, 1u, __ATOMIC_ACQ_REL, __HIP_MEMORY_SCOPE_AGENT);
    } else {
      while (__hip_atomic_load(gen, __ATOMIC_ACQUIRE, __HIP_MEMORY_SCOPE_AGENT) == g0)
        __builtin_amdgcn_s_sleep(2);
    }
  }
  __syncthreads();
  __threadfence();
}

// ---------------------------------------------------------------------------
// Kernel 1: G_x[b*S+t, n] = emb[x[b,t]] . W_ih[n,:]   (fused gather + GEMM)
// M=8192, N=2048, K=256. One block = one 16-row M tile; each wave owns 16
// contiguous N tiles processed 4-at-a-time (4 independent WMMA chains, A
// fragment shared across the 4 B tiles each k-step).
// ---------------------------------------------------------------------------
__global__ __launch_bounds__(256) void k_gx(const int* __restrict__ x,
                                            const float* __restrict__ emb,
                                            const float* __restrict__ W_ih,
                                            float* __restrict__ gx) {
  __shared__ float a_lds[16 * 260];          // padded stride: banks + 16B align
  const int tid  = threadIdx.x;
  const int lane = tid & 31, wave = tid >> 5;
  const int lo = lane & 15, hi = lane >> 4;
  const int m0 = blockIdx.x * 16;

  { // stage 16 embedding rows (gathered by token id)
    int row = tid >> 4;
    int c0  = (tid & 15) * 16;
    const float* src = emb + (long)x[m0 + row] * E_ + c0;
    float4 v0 = *(const float4*)(src);
    float4 v1 = *(const float4*)(src + 4);
    float4 v2 = *(const float4*)(src + 8);
    float4 v3 = *(const float4*)(src + 12);
    float* dst = &a_lds[row * 260 + c0];
    *(float4*)(dst)      = v0;
    *(float4*)(dst + 4)  = v1;
    *(float4*)(dst + 8)  = v2;
    *(float4*)(dst + 12) = v3;
  }
  __syncthreads();

  const float* al = &a_lds[lo * 260 + 2 * hi];
  for (int grp = 0; grp < 4; ++grp) {
    const int nt0 = wave * 16 + grp * 4;                    // first of 4 tiles
    const float* wr0 = W_ih + (long)((nt0 + 0) * 16 + lo) * E_ + 2 * hi;
    const float* wr1 = W_ih + (long)((nt0 + 1) * 16 + lo) * E_ + 2 * hi;
    const float* wr2 = W_ih + (long)((nt0 + 2) * 16 + lo) * E_ + 2 * hi;
    const float* wr3 = W_ih + (long)((nt0 + 3) * 16 + lo) * E_ + 2 * hi;
    v8f acc0 = {}, acc1 = {}, acc2 = {}, acc3 = {};
    for (int k0 = 0; k0 < E_; k0 += 4) {
      v2f a; a.x = al[k0]; a.y = al[k0 + 1];
      v2f b0; b0.x = wr0[k0]; b0.y = wr0[k0 + 1];
      v2f b1; b1.x = wr1[k0]; b1.y = wr1[k0 + 1];
      v2f b2; b2.x = wr2[k0]; b2.y = wr2[k0 + 1];
      v2f b3; b3.x = wr3[k0]; b3.y = wr3[k0 + 1];
      acc0 = WMMA_F32(a, b0, acc0);
      acc1 = WMMA_F32(a, b1, acc1);
      acc2 = WMMA_F32(a, b2, acc2);
      acc3 = WMMA_F32(a, b3, acc3);
    }
    float* dbase = gx + (long)(m0 + 8 * hi) * G4H + nt0 * 16 + lo;
#pragma unroll
    for (int r = 0; r < 8; ++r) {
      float* d = dbase + (long)r * G4H;
      d[0]  = acc0[r];
      d[16] = acc1[r];
      d[32] = acc2[r];
      d[48] = acc3[r];
    }
  }
}

// ---------------------------------------------------------------------------
// Kernel 2: persistent recurrent scan. 16 blocks (one per batch element /
// per 32-wide H chunk). F state (128KB) lives in LDS for the whole scan.
// ---------------------------------------------------------------------------
__global__ __launch_bounds__(256) void k_recur(
    const int* __restrict__ x, const float* __restrict__ W_hh,
    const float* __restrict__ b_lstm, const float* __restrict__ gx,
    const float* __restrict__ W_write, const float* __restrict__ b_write,
    const float* __restrict__ W_read, const float* __restrict__ b_read,
    const float* __restrict__ W_lin, const float* __restrict__ b_lin,
    float* __restrict__ outs, float* __restrict__ hbuf, float* __restrict__ zbuf,
    unsigned* __restrict__ bar, float* __restrict__ fraw_out) {
  extern __shared__ float L[];
  float* F      = L;                  // 32768   F[s][r][v]
  float* h_lds  = L + 32768;          // 16*516  staged h (all batches)
  float* g_lds  = L + 41024;          // 16*132  gates chunk
  float* z_g    = L + 43136;          // 512     z for owned batch
  float* wv     = L + 43648;          // 97 (pad 128)
  float* rv     = L + 43776;          // 64
  float* s_v    = L + 43840;          // 32
  float* r_v    = L + 43872;          // 32
  float* t_v    = L + 43904;          // 32
  float* q_v    = L + 43936;          // 32 (raw rv[:32])
  float* rr_v   = L + 43968;          // 32
  float* nv_v   = L + 44000;          // 32
  float* qn     = L + 44032;          // 32
  float* vpart  = L + 44064;          // 256
  float* gpart  = L + 44320;          // 256
  float* c_lds  = L + 44576;          // 512
  float* vvec   = L + 45088;          // 32
  float* gvec   = L + 45120;          // 32
  float* sc     = L + 45152;          // 16 scalars: [0]=||F||^2 [1]=beta
                                      // [2]=scale [3]=Aq [4]=sumsq [5]=mean [6]=rstd

  const int tid  = threadIdx.x;
  const int lane = tid & 31, wave = tid >> 5;
  const int lo = lane & 15, hi = lane >> 4;
  const int g  = blockIdx.x;                       // batch / h-chunk owner
  unsigned* cnt = bar;
  unsigned* gen = bar + 1;

  // ---- init state ----
  for (int i = tid; i < 32768; i += 256) F[i] = 0.f;
  c_lds[tid] = 0.f; c_lds[tid + 256] = 0.f;
  if (tid == 0) sc[0] = 0.f;
  for (int i = tid; i < H_; i += 256) hbuf[g * H_ + i] = 0.f;   // hbuf[0][g][:]
  grid_barrier(cnt, gen);

  const int qg = wave >> 1, half = wave & 1;       // gate idx, half-tile
  const int n0 = qg * H_ + g * 32 + half * 16;     // gate row base
  const float* whr = W_hh + (long)(n0 + lo) * H_ + 2 * hi;

  for (int t = 0; t < S_; ++t) {
    // ================= Phase A: gates GEMM + LSTM cell =================
    const float* hsrc = hbuf + (t & 1) * (B_ * H_);
#pragma unroll
    for (int k = 0; k < 8; ++k) {
      int el  = (tid + 256 * k) * 4;
      int b   = el >> 9, col = el & 511;
      float4 v = *(const float4*)(hsrc + el);
      *(float4*)(&h_lds[b * 516 + col]) = v;
    }
    __syncthreads();

    // split-K into 4 independent WMMA accumulation chains
    v8f acc0 = {}, acc1 = {}, acc2 = {}, acc3 = {};
    const float* hl = &h_lds[lo * 516 + 2 * hi];
    for (int k0 = 0; k0 < H_; k0 += 16) {
      v2f a0; a0.x = hl[k0 + 0];  a0.y = hl[k0 + 1];
      v2f b0; b0.x = whr[k0 + 0]; b0.y = whr[k0 + 1];
      v2f a1; a1.x = hl[k0 + 4];  a1.y = hl[k0 + 5];
      v2f b1; b1.x = whr[k0 + 4]; b1.y = whr[k0 + 5];
      v2f a2; a2.x = hl[k0 + 8];  a2.y = hl[k0 + 9];
      v2f b2; b2.x = whr[k0 + 8]; b2.y = whr[k0 + 9];
      v2f a3; a3.x = hl[k0 + 12];  a3.y = hl[k0 + 13];
      v2f b3; b3.x = whr[k0 + 12]; b3.y = whr[k0 + 13];
      acc0 = WMMA_F32(a0, b0, acc0);
      acc1 = WMMA_F32(a1, b1, acc1);
      acc2 = WMMA_F32(a2, b2, acc2);
      acc3 = WMMA_F32(a3, b3, acc3);
    }
    v8f acc = (acc0 + acc1) + (acc2 + acc3);
    {
      const int jj = qg * 32 + half * 16 + lo;
#pragma unroll
      for (int r = 0; r < 8; ++r) {
        int bb = r + 8 * hi;                        // batch row
        float gv = acc[r] + gx[((long)bb * S_ + t) * G4H + n0 + lo] + b_lstm[n0 + lo];
        g_lds[bb * 132 + jj] = gv;
      }
    }
    __syncthreads();

    for (int it = tid; it < 512; it += 256) {       // 16 batches x 32 h-elems
      int b = it >> 5, hh = it & 31;
      float gi = g_lds[b * 132 + hh];
      float gf = g_lds[b * 132 + 32 + hh];
      float gz = g_lds[b * 132 + 64 + hh];
      float go = g_lds[b * 132 + 96 + hh];
      float iv = sigm(gi), fv = sigm(gf + 1.f);
      float zg = tanhf(gz), ov = sigm(go);
      float cold = c_lds[it];
      float nc = fv * cold + iv * zg;
      float nh = ov * tanhf(nc);
      bool pad = (x[b * S_ + t] == 0);
      c_lds[it] = pad ? cold : nc;
      float hold = h_lds[b * 516 + g * 32 + hh];
      zbuf[(t & 1) * (B_ * H_) + b * H_ + g * 32 + hh] = nh;
      hbuf[((t + 1) & 1) * (B_ * H_) + b * H_ + g * 32 + hh] = pad ? hold : nh;
    }
    grid_barrier(cnt, gen);

    // ================= Phase B: fast-weight memory for batch g =================
    const float* zsrc = zbuf + (t & 1) * (B_ * H_) + g * H_;
    z_g[tid] = zsrc[tid];
    z_g[tid + 256] = zsrc[tid + 256];
    __syncthreads();

    if (tid < 97) {
      const float* w = W_write + tid * H_;
      float a = 0.f;
      for (int k = 0; k < H_; ++k) a += w[k] * z_g[k];
      wv[tid] = a + b_write[tid];
    } else if (tid >= 128 && tid < 192) {
      int j = tid - 128;
      const float* w = W_read + j * H_;
      float a = 0.f;
      for (int k = 0; k < H_; ++k) a += w[k] * z_g[k];
      rv[j] = a + b_read[j];
    }
    __syncthreads();

    if (tid < 32) {
      s_v[tid]  = tanhf(wv[tid]);
      r_v[tid]  = tanhf(wv[32 + tid]);
      t_v[tid]  = tanhf(wv[64 + tid]);
      q_v[tid]  = rv[tid];
      rr_v[tid] = tanhf(rv[32 + tid]);
    }
    if (tid == 0) sc[1] = sigm(wv[96] + 1.f);       // beta
    __syncthreads();

    { // v_v = sum_{s,r} F*s*r ; G_v = sum_{s,r} F*q*rr  (single fused pass)
      int vv = tid & 31, p = tid >> 5;
      float av = 0.f, ag = 0.f;
      for (int ss = p * 4; ss < p * 4 + 4; ++ss) {
        float sv = s_v[ss], qv = q_v[ss];
        const float* Fr = F + ss * 1024 + vv;
        for (int rr = 0; rr < 32; ++rr) {
          float fF = Fr[rr * 32];
          av += fF * sv * r_v[rr];
          ag += fF * qv * rr_v[rr];
        }
      }
      vpart[p * 32 + vv] = av;
      gpart[p * 32 + vv] = ag;
    }
    __syncthreads();
    if (tid < 32) {
      float v = 0.f, gq = 0.f;
#pragma unroll
      for (int p = 0; p < 8; ++p) { v += vpart[p * 32 + tid]; gq += gpart[p * 32 + tid]; }
      vvec[tid] = v; gvec[tid] = gq;
      nv_v[tid] = sc[1] * (t_v[tid] - v) * (1.f / 32.f);
    }
    __syncthreads();
    if (tid == 0) {
      // ||F + s r nv||^2 = ||F||^2 + 2*sum(nv*v) + (S s^2)(S r^2)(S nv^2)
      float s2 = 0, r2 = 0, sq = 0, ra = 0, n2 = 0, nd = 0;
      for (int j = 0; j < 32; ++j) {
        s2 += s_v[j] * s_v[j];   r2 += r_v[j] * r_v[j];
        sq += s_v[j] * q_v[j];   ra += r_v[j] * rr_v[j];
        n2 += nv_v[j] * nv_v[j]; nd += nv_v[j] * vvec[j];
      }
      float sumsq = sc[0] + 2.f * nd + s2 * r2 * n2;
      float fr    = sqrtf(sumsq);
      float scale = 1.f / (fmaxf(fr - 1.f, 0.f) + 1.f);
      sc[2] = scale; sc[3] = sq * ra; sc[4] = sumsq;
      if (t == S_ - 1) fraw_out[g] = fr;
    }
    __syncthreads();
    if (tid < 32) qn[tid] = sc[2] * (gvec[tid] + sc[3] * nv_v[tid]);
    __syncthreads();
    if (tid == 0) {                                  // layernorm stats
      float m = 0.f;
      for (int j = 0; j < 32; ++j) m += qn[j];
      m *= (1.f / 32.f);
      float var = 0.f;
      for (int j = 0; j < 32; ++j) { float d = qn[j] - m; var += d * d; }
      var *= (1.f / 32.f);
      sc[5] = m; sc[6] = rsqrtf(var + 1e-5f);
    }
    __syncthreads();
    if (tid < 32) qn[tid] = (qn[tid] - sc[5]) * sc[6];
    __syncthreads();

    for (int h = tid; h < H_; h += 256) {            // out = z + q @ W_lin.T + b
      const float* wl = W_lin + h * 32;
      float a = z_g[h] + b_lin[h];
#pragma unroll
      for (int j = 0; j < 32; ++j) a += wl[j] * qn[j];
      outs[((long)g * S_ + t) * H_ + h] = a;
    }

    if (x[g * S_ + t] != 0) {                        // commit F unless padded
      float scale = sc[2];
      for (int idx = tid; idx < 32768; idx += 256) {
        int ss = idx >> 10, rr = (idx >> 5) & 31, vv = idx & 31;
        F[idx] = (F[idx] + s_v[ss] * r_v[rr] * nv_v[vv]) * scale;
      }
      if (tid == 0) sc[0] = scale * scale * sc[4];
    }
    __syncthreads();
  }
}

// ---------------------------------------------------------------------------
// Kernel 3: logits = outs @ W_out.T   M=8192, N=8192, K=512.
// Each wave owns 4 N tiles (64 cols) with 4 independent WMMA chains; the A
// fragment is loaded from LDS once per k-step and reused for all 4 B tiles.
// ---------------------------------------------------------------------------
__global__ __launch_bounds__(256) void k_logits(const float* __restrict__ outs,
                                                const float* __restrict__ W_out,
                                                float* __restrict__ logits) {
  __shared__ float a_lds[16 * 516];
  const int tid  = threadIdx.x;
  const int lane = tid & 31, wave = tid >> 5;
  const int lo = lane & 15, hi = lane >> 4;
  const int m0    = blockIdx.x * 16;
  const int nbase = blockIdx.y * 512;

#pragma unroll
  for (int k = 0; k < 8; ++k) {
    int el = (tid + 256 * k) * 4;
    int rw = el >> 9, col = el & 511;
    float4 v = *(const float4*)(outs + (long)(m0 + rw) * H_ + col);
    *(float4*)(&a_lds[rw * 516 + col]) = v;
  }
  __syncthreads();

  const float* al = &a_lds[lo * 516 + 2 * hi];
  const int n0 = nbase + wave * 64;
  const float* wr0 = W_out + (long)(n0 + 0  + lo) * H_ + 2 * hi;
  const float* wr1 = W_out + (long)(n0 + 16 + lo) * H_ + 2 * hi;
  const float* wr2 = W_out + (long)(n0 + 32 + lo) * H_ + 2 * hi;
  const float* wr3 = W_out + (long)(n0 + 48 + lo) * H_ + 2 * hi;
  v8f acc0 = {}, acc1 = {}, acc2 = {}, acc3 = {};
  for (int k0 = 0; k0 < H_; k0 += 4) {
    v2f a; a.x = al[k0]; a.y = al[k0 + 1];
    v2f b0; b0.x = wr0[k0]; b0.y = wr0[k0 + 1];
    v2f b1; b1.x = wr1[k0]; b1.y = wr1[k0 + 1];
    v2f b2; b2.x = wr2[k0]; b2.y = wr2[k0 + 1];
    v2f b3; b3.x = wr3[k0]; b3.y = wr3[k0 + 1];
    acc0 = WMMA_F32(a, b0, acc0);
    acc1 = WMMA_F32(a, b1, acc1);
    acc2 = WMMA_F32(a, b2, acc2);
    acc3 = WMMA_F32(a, b3, acc3);
  }
  float* dbase = logits + (long)(m0 + 8 * hi) * V_ + n0 + lo;
#pragma unroll
  for (int r = 0; r < 8; ++r) {
    float* d = dbase + (long)r * V_;
    d[0]  = acc0[r];
    d[16] = acc1[r];
    d[32] = acc2[r];
    d[48] = acc3[r];
  }
}

// ---------------------------------------------------------------------------
extern "C" void kernel_launch(void* const* d_in, const int* in_sizes, int n_in,
                              void* d_out, int out_size, void* d_ws, size_t ws_size,
                              hipStream_t stream) {
  (void)in_sizes; (void)n_in; (void)out_size; (void)ws_size;
  const int*   x       = (const int*)  d_in[0];
  // d_in[1] = seq_len (unused by the reference math)
  const float* emb     = (const float*)d_in[2];
  const float* W_ih    = (const float*)d_in[3];
  const float* W_hh    = (const float*)d_in[4];
  const float* b_lstm  = (const float*)d_in[5];
  const float* W_write = (const float*)d_in[6];
  const float* b_write = (const float*)d_in[7];
  const float* W_read  = (const float*)d_in[8];
  const float* b_read  = (const float*)d_in[9];
  const float* W_lin   = (const float*)d_in[10];
  const float* b_lin   = (const float*)d_in[11];
  const float* W_out   = (const float*)d_in[12];

  float* ws   = (float*)d_ws;
  float* gx   = ws;                   // 8192*2048 = 16,777,216 f32 (64 MB)
  float* outsb= ws + 16777216;        // 8192*512  =  4,194,304 f32 (16 MB)
  float* hbuf = ws + 20971520;        // 2*16*512  = 16,384 f32
  float* zbuf = ws + 20987904;        // 2*16*512  = 16,384 f32
  unsigned* bar = (unsigned*)(ws + 21004288);

  float* logits   = (float*)d_out;
  float* fraw_out = logits + (long)B_ * S_ * V_;

  hipMemsetAsync(bar, 0, 64, stream);                       // reset grid barrier
  k_gx    <<<512, 256, 0, stream>>>(x, emb, W_ih, gx);
  k_recur <<<NBLK, 256, 45168 * 4, stream>>>(x, W_hh, b_lstm, gx,
                                             W_write, b_write, W_read, b_read,
                                             W_lin, b_lin, outsb, hbuf, zbuf,
                                             bar, fraw_out);
  k_logits<<<dim3(512, 16), 256, 0, stream>>>(outsb, W_out, logits);
}